// Earl_82411832476066
// MI455X (gfx1250) — compile-verified
//
#include <hip/hip_runtime.h>
#include <hip/hip_bf16.h>

// ---------------------------------------------------------------------------
// Problem constants (match reference setup_inputs()).
// ---------------------------------------------------------------------------
constexpr int kD  = 768;         // embedding dim
constexpr int kG3 = 3 * kD;      // 2304 (3 gates)
constexpr int kS  = 2048;        // seeds
constexpr int kE  = 32768;       // edges
constexpr int kN  = 100000;      // nodes
constexpr int kR  = 1000;        // relations
constexpr int kSTRIDE = kE + 1;  // timestamp stride (32769)

typedef __attribute__((ext_vector_type(2))) float v2f;
typedef __attribute__((ext_vector_type(8))) float v8f;

__device__ __forceinline__ float sigmoidf_(float x) {
    return 1.0f / (1.0f + expf(-x));
}

// ---------------------------------------------------------------------------
// Scatter bookkeeping kernels (timestamps / last-write-wins emulation)
// ---------------------------------------------------------------------------
__global__ void init_kernel(int* __restrict__ best, int* __restrict__ seedpos) {
    int i = blockIdx.x * blockDim.x + threadIdx.x;
    if (i < kN) { best[i] = -1; seedpos[i] = -1; }
}

__global__ void seed_scatter_kernel(const int* __restrict__ seed_ent,
                                    int* __restrict__ seedpos,
                                    int* __restrict__ best) {
    int s = blockIdx.x * blockDim.x + threadIdx.x;
    if (s < kS) {
        int node = seed_ent[s];
        seedpos[node] = s;
        atomicMax(&best[node], s * kSTRIDE);
    }
}

__global__ void edge_ts_kernel(const int* __restrict__ heads,
                               const int* __restrict__ tails,
                               const int* __restrict__ seedpos,
                               int* __restrict__ best) {
    int e = blockIdx.x * blockDim.x + threadIdx.x;
    if (e < kE) {
        long t = (long)seedpos[heads[e]] * kSTRIDE + 1 + e;  // may be negative
        atomicMax(&best[tails[e]], (int)t);
    }
}

// ---------------------------------------------------------------------------
// gi0[j] = b_ih[j] + dot(encoder_output, W_ih[j,:])  (x0 identical for all S)
// ---------------------------------------------------------------------------
__global__ void gi0_kernel(const float* __restrict__ enc,
                           const float* __restrict__ W_ih,
                           const float* __restrict__ b_ih,
                           float* __restrict__ gi0) {
    int j = blockIdx.x * blockDim.x + threadIdx.x;
    if (j < kG3) {
        const float* w = W_ih + (size_t)j * kD;
        float s = b_ih[j];
        for (int k = 0; k < kD; ++k) s += enc[k] * w[k];
        gi0[j] = s;
    }
}

// ---------------------------------------------------------------------------
// Generic NT GEMM: C[m,n] = act(sum_k A[m,k]*W[n,k] + bias[n])
// One wave32 -> one 16x16 tile via V_WMMA_F32_16X16X4_F32.
// Block = 8 waves = 32(M) x 64(N) output.
// ---------------------------------------------------------------------------
__global__ __launch_bounds__(256)
void gemm_nt_kernel(const float* __restrict__ A, const float* __restrict__ W,
                    const float* __restrict__ bias, float* __restrict__ C,
                    int M, int Nout, int act) {
    int tid  = threadIdx.x;
    int wave = tid >> 5, lane = tid & 31;
    int l = lane & 15, h = lane >> 4;
    int m0 = blockIdx.y * 32 + (wave >> 2) * 16;
    int n0 = blockIdx.x * 64 + (wave & 3) * 16;

    int arow = m0 + l; if (arow >= M) arow = M - 1;   // clamp (keep EXEC all-1s)
    const float* ap = A + (size_t)arow * kD + 2 * h;
    const float* bp = W + (size_t)(n0 + l) * kD + 2 * h;

    v8f acc = {0.f,0.f,0.f,0.f,0.f,0.f,0.f,0.f};
#pragma unroll 8
    for (int k = 0; k < kD; k += 4) {
        v2f a = *(const v2f*)(ap + k);
        v2f b = *(const v2f*)(bp + k);
        acc = __builtin_amdgcn_wmma_f32_16x16x4_f32(false, a, false, b,
                                                    (short)0, acc, false, false);
    }
    int n = n0 + l;
    float bn = bias[n];
#pragma unroll
    for (int v = 0; v < 8; ++v) {
        int m = m0 + v + 8 * h;
        if (m < M) {
            float x = acc[v] + bn;
            if (act) x = tanhf(x);
            C[(size_t)m * Nout + n] = x;
        }
    }
}

// ---------------------------------------------------------------------------
// r0 = GRUCell(encoder_output, sub): gh = sub @ W_hh^T (+b_hh), gi0 broadcast.
// 3 gate accumulators per tile; one A fragment feeds three WMMAs per K-step.
// ---------------------------------------------------------------------------
__global__ __launch_bounds__(256)
void r0_kernel(const float* __restrict__ sub, const float* __restrict__ W_hh,
               const float* __restrict__ b_hh, const float* __restrict__ gi0,
               float* __restrict__ r0) {
    int tid  = threadIdx.x;
    int wave = tid >> 5, lane = tid & 31;
    int l = lane & 15, h = lane >> 4;
    int m0 = blockIdx.y * 32 + (wave >> 2) * 16;     // M = 2048
    int n0 = blockIdx.x * 64 + (wave & 3) * 16;      // N = 768

    const float* ap = sub + (size_t)(m0 + l) * kD + 2 * h;
    const float* bR = W_hh + (size_t)(n0 + l) * kD + 2 * h;
    const float* bZ = W_hh + (size_t)(kD + n0 + l) * kD + 2 * h;
    const float* bN = W_hh + (size_t)(2 * kD + n0 + l) * kD + 2 * h;

    v8f aR = {0.f,0.f,0.f,0.f,0.f,0.f,0.f,0.f};
    v8f aZ = aR, aN = aR;
#pragma unroll 4
    for (int k = 0; k < kD; k += 4) {
        v2f a  = *(const v2f*)(ap + k);
        v2f br = *(const v2f*)(bR + k);
        v2f bz = *(const v2f*)(bZ + k);
        v2f bn = *(const v2f*)(bN + k);
        aR = __builtin_amdgcn_wmma_f32_16x16x4_f32(false, a, false, br, (short)0, aR, false, false);
        aZ = __builtin_amdgcn_wmma_f32_16x16x4_f32(false, a, false, bz, (short)0, aZ, false, false);
        aN = __builtin_amdgcn_wmma_f32_16x16x4_f32(false, a, false, bn, (short)0, aN, false, false);
    }
    int n = n0 + l;
    float bhR = b_hh[n], bhZ = b_hh[kD + n], bhN = b_hh[2 * kD + n];
    float giR = gi0[n],  giZ = gi0[kD + n],  giN = gi0[2 * kD + n];
#pragma unroll
    for (int v = 0; v < 8; ++v) {
        int m = m0 + v + 8 * h;
        float r  = sigmoidf_(giR + aR[v] + bhR);
        float z  = sigmoidf_(giZ + aZ[v] + bhZ);
        float nn = tanhf(giN + r * (aN[v] + bhN));
        float hv = sub[(size_t)m * kD + n];
        r0[(size_t)m * kD + n] = (1.f - z) * nn + z * hv;
    }
}

// ---------------------------------------------------------------------------
// Fused edge kernel: 16 edges per block.
//   phase1: rj = GRUCell(r0[head], rel[type])  -> rj tile lives in LDS only
//   phase2: obj = tanh(rj @ W_obj^T + b_obj), scatter winning rows to table.
// gh_e comes from the precomputed ghr = rel_emb @ W_hh^T + b_hh (R rows).
// ---------------------------------------------------------------------------
__global__ __launch_bounds__(256)
void edge_kernel(const float* __restrict__ r0, const float* __restrict__ ghr,
                 const float* __restrict__ rel_emb,
                 const float* __restrict__ W_ih, const float* __restrict__ b_ih,
                 const float* __restrict__ W_obj, const float* __restrict__ b_obj,
                 const int* __restrict__ edge_heads, const int* __restrict__ edge_tails,
                 const int* __restrict__ edge_type, const int* __restrict__ seedpos,
                 const int* __restrict__ best, float* __restrict__ table) {
    __shared__ float rj[16][kD + 4];                 // +4 pad: conflict-free frags
    __shared__ int sh_head[16], sh_type[16], sh_tail[16], sh_win[16];

    int tid = threadIdx.x;
    int e0  = blockIdx.x * 16;
    if (tid < 16) {
        int e  = e0 + tid;
        int hd = edge_heads[e];
        int tl = edge_tails[e];
        sh_head[tid] = hd;
        sh_type[tid] = edge_type[e];
        sh_tail[tid] = tl;
        long t = (long)seedpos[hd] * kSTRIDE + 1 + e;
        sh_win[tid] = (t >= 0 && best[tl] == (int)t) ? 1 : 0;
    }
    __syncthreads();

    int wave = tid >> 5, lane = tid & 31;
    int l = lane & 15, h = lane >> 4;
    const float* ap = r0 + (size_t)sh_head[l] * kD + 2 * h;  // gathered A rows

    // ---- phase 1: gi GEMM + GRU combine -> rj (LDS) ----
    for (int ct = 0; ct < 6; ++ct) {                 // 48 N-tiles / 8 waves
        int n0 = (wave * 6 + ct) * 16;
        const float* bR = W_ih + (size_t)(n0 + l) * kD + 2 * h;
        const float* bZ = W_ih + (size_t)(kD + n0 + l) * kD + 2 * h;
        const float* bN = W_ih + (size_t)(2 * kD + n0 + l) * kD + 2 * h;
        v8f aR = {0.f,0.f,0.f,0.f,0.f,0.f,0.f,0.f};
        v8f aZ = aR, aN = aR;
#pragma unroll 4
        for (int k = 0; k < kD; k += 4) {
            v2f a  = *(const v2f*)(ap + k);
            v2f br = *(const v2f*)(bR + k);
            v2f bz = *(const v2f*)(bZ + k);
            v2f bn = *(const v2f*)(bN + k);
            aR = __builtin_amdgcn_wmma_f32_16x16x4_f32(false, a, false, br, (short)0, aR, false, false);
            aZ = __builtin_amdgcn_wmma_f32_16x16x4_f32(false, a, false, bz, (short)0, aZ, false, false);
            aN = __builtin_amdgcn_wmma_f32_16x16x4_f32(false, a, false, bn, (short)0, aN, false, false);
        }
        int n = n0 + l;
        float biR = b_ih[n], biZ = b_ih[kD + n], biN = b_ih[2 * kD + n];
#pragma unroll
        for (int v = 0; v < 8; ++v) {
            int m  = v + 8 * h;
            int ty = sh_type[m];
            const float* gh = ghr + (size_t)ty * kG3;   // includes b_hh
            float r  = sigmoidf_(aR[v] + biR + gh[n]);
            float z  = sigmoidf_(aZ[v] + biZ + gh[kD + n]);
            float nn = tanhf(aN[v] + biN + r * gh[2 * kD + n]);
            float hv = rel_emb[(size_t)ty * kD + n];
            rj[m][n] = (1.f - z) * nn + z * hv;
        }
    }
    __syncthreads();

    // ---- phase 2: obj = tanh(rj @ W_obj^T + b_obj), scatter winners ----
    const float* ap2 = &rj[l][2 * h];
    for (int ct = 0; ct < 6; ++ct) {
        int n0 = (wave * 6 + ct) * 16;
        const float* bp = W_obj + (size_t)(n0 + l) * kD + 2 * h;
        v8f acc = {0.f,0.f,0.f,0.f,0.f,0.f,0.f,0.f};
#pragma unroll 8
        for (int k = 0; k < kD; k += 4) {
            v2f a = *(const v2f*)(ap2 + k);            // ds_load from LDS
            v2f b = *(const v2f*)(bp + k);
            acc = __builtin_amdgcn_wmma_f32_16x16x4_f32(false, a, false, b,
                                                        (short)0, acc, false, false);
        }
        int n = n0 + l;
        float bo = b_obj[n];
#pragma unroll
        for (int v = 0; v < 8; ++v) {
            int m = v + 8 * h;
            if (sh_win[m])
                table[(size_t)sh_tail[m] * kD + n] = tanhf(acc[v] + bo);
        }
    }
}

// ---------------------------------------------------------------------------
// Seed rows that win last-write go to table; final gather row-by-row.
// ---------------------------------------------------------------------------
__global__ void seed_table_kernel(const int* __restrict__ seed_ent,
                                  const int* __restrict__ best,
                                  const float* __restrict__ sub,
                                  float* __restrict__ table) {
    int s = blockIdx.x;
    int node = seed_ent[s];
    if (best[node] != s * kSTRIDE) return;
    const float4* src = (const float4*)(sub + (size_t)s * kD);
    float4* dst = (float4*)(table + (size_t)node * kD);
    for (int i = threadIdx.x; i < kD / 4; i += blockDim.x) dst[i] = src[i];
}

__global__ void gather_kernel(const int* __restrict__ node2node,
                              const int* __restrict__ old_ndata,
                              const int* __restrict__ best,
                              const float* __restrict__ table,
                              const float* __restrict__ defaults,
                              float* __restrict__ out) {
    int i  = blockIdx.x;
    int id = node2node[old_ndata[i]];
    bool wrote = best[id] >= 0;
    const float4* src = wrote ? (const float4*)(table + (size_t)id * kD)
                              : (const float4*)(defaults + (size_t)i * kD);
    float4* dst = (float4*)(out + (size_t)i * kD);
    for (int j = threadIdx.x; j < kD / 4; j += blockDim.x) dst[j] = src[j];
}

// ---------------------------------------------------------------------------
// Host-side launcher
// ---------------------------------------------------------------------------
extern "C" void kernel_launch(void* const* d_in, const int* in_sizes, int n_in,
                              void* d_out, int out_size, void* d_ws, size_t ws_size,
                              hipStream_t stream) {
    const float* enc   = (const float*)d_in[0];
    const float* smask = (const float*)d_in[1];
    const int*   seedE = (const int*)d_in[2];
    const int*   eh    = (const int*)d_in[3];
    const int*   etl   = (const int*)d_in[4];
    const int*   ety   = (const int*)d_in[5];
    const int*   n2n   = (const int*)d_in[6];
    const int*   oldn  = (const int*)d_in[7];
    const float* rel   = (const float*)d_in[8];
    const float* Wsub  = (const float*)d_in[9];
    const float* bsub  = (const float*)d_in[10];
    const float* Wobj  = (const float*)d_in[11];
    const float* bobj  = (const float*)d_in[12];
    const float* Wih   = (const float*)d_in[13];
    const float* Whh   = (const float*)d_in[14];
    const float* bih   = (const float*)d_in[15];
    const float* bhh   = (const float*)d_in[16];
    const float* defn  = (const float*)d_in[17];
    float* out = (float*)d_out;
    (void)in_sizes; (void)n_in; (void)out_size; (void)ws_size;

    char*  ws  = (char*)d_ws;
    size_t off = 0;
    auto carve = [&](size_t bytes) -> char* {
        char* p = ws + off;
        off += (bytes + 255) & ~(size_t)255;
        return p;
    };
    float* sub    = (float*)carve((size_t)kS * kD * 4);
    float* r0     = (float*)carve((size_t)kS * kD * 4);
    float* gi0    = (float*)carve((size_t)kG3 * 4);
    float* ghr    = (float*)carve((size_t)kR * kG3 * 4);
    float* table  = (float*)carve((size_t)kN * kD * 4);
    int*   best   = (int*)carve((size_t)kN * 4);
    int*   seedps = (int*)carve((size_t)kN * 4);

    // Timestamp / winner machinery
    init_kernel<<<(kN + 255) / 256, 256, 0, stream>>>(best, seedps);
    seed_scatter_kernel<<<(kS + 255) / 256, 256, 0, stream>>>(seedE, seedps, best);
    edge_ts_kernel<<<(kE + 255) / 256, 256, 0, stream>>>(eh, etl, seedps, best);

    // gi0 (encoder gate preactivations, shared by all seeds)
    gi0_kernel<<<(kG3 + 255) / 256, 256, 0, stream>>>(enc, Wih, bih, gi0);

    // sub = tanh(sample_mask @ W_sub^T + b_sub)  [2048 x 768]
    gemm_nt_kernel<<<dim3(kD / 64, kS / 32), 256, 0, stream>>>(
        smask, Wsub, bsub, sub, kS, kD, 1);

    // ghr = rel_emb @ W_hh^T + b_hh  [1000 x 2304] (hoisted out of edge loop)
    gemm_nt_kernel<<<dim3(kG3 / 64, (kR + 31) / 32), 256, 0, stream>>>(
        rel, Whh, bhh, ghr, kR, kG3, 0);

    // r0 = GRUCell(encoder, sub)  [2048 x 768]
    r0_kernel<<<dim3(kD / 64, kS / 32), 256, 0, stream>>>(sub, Whh, bhh, gi0, r0);

    // Fused per-edge GRU + obj projection + winner scatter
    edge_kernel<<<kE / 16, 256, 0, stream>>>(r0, ghr, rel, Wih, bih, Wobj, bobj,
                                             eh, etl, ety, seedps, best, table);

    // Winning seed rows
    seed_table_kernel<<<kS, 192, 0, stream>>>(seedE, best, sub, table);

    // Final gather
    gather_kernel<<<kN, 192, 0, stream>>>(n2n, oldn, best, table, defn, out);
}